// ConvAttention_70798240907435
// MI455X (gfx1250) — compile-verified
//
#include <hip/hip_runtime.h>
#include <hip/hip_bf16.h>
#include <math.h>

typedef float v2f __attribute__((ext_vector_type(2)));
typedef float v8f __attribute__((ext_vector_type(8)));
typedef int   v4i_t __attribute__((vector_size(16)));
typedef __attribute__((address_space(1))) v4i_t g_v4i;   // global int4
typedef __attribute__((address_space(3))) v4i_t l_v4i;   // LDS int4

// Problem constants (from reference): b=8, c=512, h=w=64, oc=3c, heads=8
constexpr int BB  = 8;
constexpr int CC  = 512;
constexpr int LL  = 4096;        // h*w
constexpr int OCC = 1536;        // 3*c
constexpr int HEADS = 8;
constexpr int HD  = 64;          // c / heads

constexpr size_t XO_SZ = (size_t)BB * CC * LL;   // 16,777,216
constexpr size_t KOFF  = XO_SZ;
constexpr size_t VOFF  = 2 * XO_SZ;

// workspace layout (float offsets)
constexpr size_t GL_OFF   = 0;                   // 4096
constexpr size_t GIN_OFF  = 4096;                // 4096
constexpr size_t GOUT_OFF = 8192;                // 12288
constexpr size_t INVQ_OFF = 20480;               // 4096
constexpr size_t INVK_OFF = 24576;               // 4096
constexpr size_t DYNW_OFF = 28672;               // 6,291,456  (~25.3 MB total)

#if defined(__gfx1250__) && __has_builtin(__builtin_amdgcn_global_load_async_to_lds_b128)
#define HAS_ASYNC_LDS 1
#endif

// 16-byte global -> LDS copy; async (ASYNCcnt, no VGPR bounce) when available.
static __device__ __forceinline__ void cp16(const float* g, float* l) {
#ifdef HAS_ASYNC_LDS
  __builtin_amdgcn_global_load_async_to_lds_b128((g_v4i*)(g), (l_v4i*)(l), 0, 0);
#else
  *(float4*)l = *(const float4*)g;
#endif
}

static __device__ __forceinline__ void wait_async0() {
#ifdef HAS_ASYNC_LDS
#if __has_builtin(__builtin_amdgcn_s_wait_asynccnt)
  __builtin_amdgcn_s_wait_asynccnt(0);
#else
  asm volatile("s_wait_asynccnt 0x0" ::: "memory");
#endif
#endif
}

static __device__ __forceinline__ v8f vzero8() {
  v8f z;
#pragma unroll
  for (int i = 0; i < 8; ++i) z[i] = 0.0f;
  return z;
}

static __device__ __forceinline__ v8f wmma_f32_k4(v2f a, v2f b, v8f c) {
#if defined(__gfx1250__) && __has_builtin(__builtin_amdgcn_wmma_f32_16x16x4_f32)
  // D = A(16x4) * B(4x16) + C(16x16), fp32 throughout; emits v_wmma_f32_16x16x4_f32
  return __builtin_amdgcn_wmma_f32_16x16x4_f32(false, a, false, b, (short)0, c, false, false);
#else
  c[0] += a.x * b.x;  // placeholder fallback so the TU still compiles
  return c;
#endif
}

// ---------------- Kernel 1: per-(b,c) max over L (AdaptiveMaxPool) ----------------
__global__ __launch_bounds__(256) void colmax_k(const float* __restrict__ x,
                                                float* __restrict__ gl) {
  __shared__ float s[256];
  const int bc = blockIdx.x;
  const float* row = x + (size_t)bc * LL;
  float m = -INFINITY;
  for (int i = threadIdx.x; i < LL; i += 256) m = fmaxf(m, row[i]);
  s[threadIdx.x] = m;
  __syncthreads();
  for (int o = 128; o > 0; o >>= 1) {
    if (threadIdx.x < o) s[threadIdx.x] = fmaxf(s[threadIdx.x], s[threadIdx.x + o]);
    __syncthreads();
  }
  if (threadIdx.x == 0) gl[bc] = s[0];
}

// ---------------- Kernel 2: gate MLPs (ce/gd, ci/gd2) ----------------
__global__ __launch_bounds__(256) void gates_k(const float* __restrict__ gl,
                                               const float* __restrict__ ce_w,
                                               const float* __restrict__ ci_w,
                                               const float* __restrict__ gd_w,
                                               const float* __restrict__ gd2_w,
                                               float* __restrict__ gate_in,
                                               float* __restrict__ gate_out) {
  __shared__ float rce[CC];
  const int b = blockIdx.x;
  const float ce = ce_w[0], gd = gd_w[0], gd2 = gd2_w[0];
  for (int c = threadIdx.x; c < CC; c += 256) {
    float r = fmaxf(gl[b * CC + c] * ce, 0.0f);
    rce[c] = r;
    gate_in[b * CC + c] = r * gd;
  }
  __syncthreads();
  for (int o = threadIdx.x; o < OCC; o += 256) {
    const int n = o / 24, j = o % 24;     // ocv reshape: o = n*24 + j
    float s = 0.0f;
#pragma unroll
    for (int g = 0; g < 8; ++g) s += rce[n * 8 + g] * ci_w[j * 8 + g];
    gate_out[b * OCC + o] = fmaxf(s, 0.0f) * gd2;
  }
}

// ---------------- Kernel 3: materialize dynamic weight ----------------
__global__ __launch_bounds__(256) void dynw_k(const float* __restrict__ W,
                                              const float* __restrict__ gate_in,
                                              const float* __restrict__ gate_out,
                                              float* __restrict__ dynw) {
  const int total = BB * OCC * CC;
  for (int i = blockIdx.x * 256 + threadIdx.x; i < total; i += gridDim.x * 256) {
    const int ci = i & (CC - 1);
    const int o  = (i / CC) % OCC;
    const int b  = i / (CC * OCC);
    const float z  = gate_in[b * CC + ci] + gate_out[b * OCC + o];
    const float sg = 1.0f / (1.0f + __expf(-z));
    dynw[i] = sg * W[o * CC + ci];
  }
}

// ---------------- Kernel 4: batched GEMM qkv[b] = dynw[b] (1536x512) @ x[b] (512x4096) ----------------
// block tile 256(M)x128(N), KC=32, 8 waves as 4x2 grid, wave tile 64x64 (4x4 WMMA tiles).
// A tile staged with async global->LDS b128 (stride 36: 16B-aligned rows, conflict-free frag reads);
// B tile stored transposed with odd stride 33 (conflict-free).
constexpr int APAD = 36;
__global__ __launch_bounds__(256) void gemm_qkv_k(const float* __restrict__ dynw,
                                                  const float* __restrict__ x,
                                                  float* __restrict__ out) {
  __shared__ float Alds[256 * APAD];   // 36.9 KB
  __shared__ float Blds[128 * 33];     // 16.9 KB
  const int tid = threadIdx.x;
  const int lane = tid & 31, wave = tid >> 5;
  const int hi = lane >> 4, ln = lane & 15;
  const int n0 = blockIdx.x * 128;
  const int m0 = blockIdx.y * 256;
  const int b  = blockIdx.z;
  const int waveM = wave >> 1, waveN = wave & 1;

  const float* Ag = dynw + (size_t)b * OCC * CC;
  const float* Bg = x    + (size_t)b * CC  * LL;

  v8f acc[4][4];
#pragma unroll
  for (int mi = 0; mi < 4; ++mi)
#pragma unroll
    for (int ni = 0; ni < 4; ++ni) acc[mi][ni] = vzero8();

  for (int k0 = 0; k0 < CC; k0 += 32) {
    // A tile 256x32: contiguous 16B chunks, async -> LDS (8 per thread)
    for (int i = tid; i < 2048; i += 256) {
      const int row = i >> 3, c4 = (i & 7) * 4;
      cp16(&Ag[(m0 + row) * CC + k0 + c4], &Alds[row * APAD + c4]);
    }
    // B tile transposed -> Blds[n][kk] (float4 global reads, scattered b32 LDS stores)
    for (int i = tid; i < 1024; i += 256) {
      const int kk = i >> 5, n4 = (i & 31) * 4;
      const float4 v4 = *(const float4*)&Bg[(size_t)(k0 + kk) * LL + n0 + n4];
      Blds[(n4 + 0) * 33 + kk] = v4.x;
      Blds[(n4 + 1) * 33 + kk] = v4.y;
      Blds[(n4 + 2) * 33 + kk] = v4.z;
      Blds[(n4 + 3) * 33 + kk] = v4.w;
    }
    wait_async0();
    __syncthreads();
#pragma unroll
    for (int kk = 0; kk < 32; kk += 4) {
      v2f av[4], bv[4];
#pragma unroll
      for (int mi = 0; mi < 4; ++mi) {
        const int r = (waveM * 64 + mi * 16 + ln) * APAD + kk + 2 * hi;
        av[mi].x = Alds[r]; av[mi].y = Alds[r + 1];
      }
#pragma unroll
      for (int ni = 0; ni < 4; ++ni) {
        const int r = (waveN * 64 + ni * 16 + ln) * 33 + kk + 2 * hi;
        bv[ni].x = Blds[r]; bv[ni].y = Blds[r + 1];
      }
#pragma unroll
      for (int mi = 0; mi < 4; ++mi)
#pragma unroll
        for (int ni = 0; ni < 4; ++ni)
          acc[mi][ni] = wmma_f32_k4(av[mi], bv[ni], acc[mi][ni]);
    }
    __syncthreads();
  }

  // epilogue: route q -> xo region (temp), k/v -> final output slices
#pragma unroll
  for (int mi = 0; mi < 4; ++mi) {
#pragma unroll
    for (int ni = 0; ni < 4; ++ni) {
      const int col = n0 + waveN * 64 + ni * 16 + ln;
#pragma unroll
      for (int r = 0; r < 8; ++r) {
        const int o = m0 + waveM * 64 + mi * 16 + r + 8 * hi;
        const float v = acc[mi][ni][r];
        size_t off;
        if (o < CC)            off = ((size_t)b * CC + o) * LL + col;                  // q (temp in xo)
        else if (o < 2 * CC)   off = KOFF + ((size_t)b * CC + (o - CC)) * LL + col;    // k
        else                   off = VOFF + ((size_t)b * CC + (o - 2 * CC)) * LL + col;// v
        out[off] = v;
      }
    }
  }
}

// ---------------- Kernel 5: inverse L2 row norms for q and k ----------------
__global__ __launch_bounds__(256) void rownorm_k(const float* __restrict__ out,
                                                 float* __restrict__ invq,
                                                 float* __restrict__ invk) {
  __shared__ float s[256];
  const int r = blockIdx.x;
  const float* row;
  float* dst;
  int idx;
  if (r < BB * CC) { idx = r;           row = out + (size_t)idx * LL;        dst = invq; }
  else             { idx = r - BB * CC; row = out + KOFF + (size_t)idx * LL; dst = invk; }
  float ss = 0.0f;
  for (int i = threadIdx.x; i < LL; i += 256) { const float v = row[i]; ss += v * v; }
  s[threadIdx.x] = ss;
  __syncthreads();
  for (int o = 128; o > 0; o >>= 1) {
    if (threadIdx.x < o) s[threadIdx.x] += s[threadIdx.x + o];
    __syncthreads();
  }
  if (threadIdx.x == 0) dst[idx] = 1.0f / fmaxf(sqrtf(s[0]), 1e-12f);
}

// ---------------- Kernel 6: fused XCA attention per (b,h): S = softmax(qn.kn^T * aw); xo = S @ v ----------------
constexpr int QPAD = 68;   // 16B-aligned rows for async staging; 4*ln mod 64 distinct -> conflict-free
__global__ __launch_bounds__(256) void attn_xo_k(const float* __restrict__ invq,
                                                 const float* __restrict__ invk,
                                                 const float* __restrict__ attn_w,
                                                 float* __restrict__ out) {
  __shared__ float Qs[64 * QPAD];
  __shared__ float Ks[64 * QPAD];
  __shared__ float Ss[64 * 65];
  const int tid = threadIdx.x, lane = tid & 31, wave = tid >> 5;
  const int hi = lane >> 4, ln = lane & 15;
  const int b = blockIdx.x >> 3, h = blockIdx.x & 7;
  const size_t rowbase = ((size_t)b * CC + h * HD) * LL;
  const float* qp = out + rowbase;          // q lives temporarily in xo region
  const float* kp = out + KOFF + rowbase;
  const float* vp = out + VOFF + rowbase;

  // 16 (4x4) C-tiles over 8 waves: wave handles tiles (mt0, ni) and (mt0+2, ni)
  const int mt0 = wave >> 2, ni = wave & 3;
  v8f acc0 = vzero8(), acc1 = vzero8();

  for (int kc = 0; kc < LL; kc += 64) {
    // stage 64x64 q and k tiles via async b128 copies (16B chunks)
    for (int i = tid; i < 1024; i += 256) {
      const int row = i >> 4, c4 = (i & 15) * 4;
      cp16(&qp[(size_t)row * LL + kc + c4], &Qs[row * QPAD + c4]);
      cp16(&kp[(size_t)row * LL + kc + c4], &Ks[row * QPAD + c4]);
    }
    wait_async0();
    __syncthreads();
#pragma unroll
    for (int ks = 0; ks < 64; ks += 4) {
      v2f a0, a1, bb;
      const int ra0 = (mt0 * 16 + ln) * QPAD + ks + 2 * hi;
      const int ra1 = ((mt0 + 2) * 16 + ln) * QPAD + ks + 2 * hi;
      const int rb  = (ni * 16 + ln) * QPAD + ks + 2 * hi;
      a0.x = Qs[ra0]; a0.y = Qs[ra0 + 1];
      a1.x = Qs[ra1]; a1.y = Qs[ra1 + 1];
      bb.x = Ks[rb];  bb.y = Ks[rb + 1];
      acc0 = wmma_f32_k4(a0, bb, acc0);
      acc1 = wmma_f32_k4(a1, bb, acc1);
    }
    __syncthreads();
  }

  // scale by invq*invk*attn_w, deposit into Ss
  const int cbase = b * CC + h * HD;
#pragma unroll
  for (int r = 0; r < 8; ++r) {
    const int m0r = mt0 * 16 + r + 8 * hi;
    const int m1r = (mt0 + 2) * 16 + r + 8 * hi;
    const int n = ni * 16 + ln;
    const float ikn = invk[cbase + n];
    Ss[m0r * 65 + n] = acc0[r] * invq[cbase + m0r] * ikn * attn_w[(h * HD + m0r) * HD + n];
    Ss[m1r * 65 + n] = acc1[r] * invq[cbase + m1r] * ikn * attn_w[(h * HD + m1r) * HD + n];
  }
  __syncthreads();

  // softmax over rows (64x64, one thread per row)
  if (tid < 64) {
    float mx = -INFINITY;
    for (int j = 0; j < 64; ++j) mx = fmaxf(mx, Ss[tid * 65 + j]);
    float sm = 0.0f;
    for (int j = 0; j < 64; ++j) { const float e = __expf(Ss[tid * 65 + j] - mx); Ss[tid * 65 + j] = e; sm += e; }
    const float inv = 1.0f / sm;
    for (int j = 0; j < 64; ++j) Ss[tid * 65 + j] *= inv;
  }
  __syncthreads();

  // xo = S (64x64) @ v (64x4096); overwrite the consumed q slice with xo
  float* xop = out + rowbase;
  for (int nt = wave; nt < LL / 16; nt += 8) {
    const int col = nt * 16 + ln;
    v2f bfr[16];
#pragma unroll
    for (int ks = 0; ks < 16; ++ks) {
      const int d = ks * 4 + 2 * hi;
      bfr[ks].x = vp[(size_t)d * LL + col];
      bfr[ks].y = vp[(size_t)(d + 1) * LL + col];
    }
#pragma unroll
    for (int mt = 0; mt < 4; ++mt) {
      v8f acc = vzero8();
#pragma unroll
      for (int ks = 0; ks < 16; ++ks) {
        v2f a;
        const int ra = (mt * 16 + ln) * 65 + ks * 4 + 2 * hi;
        a.x = Ss[ra]; a.y = Ss[ra + 1];
        acc = wmma_f32_k4(a, bfr[ks], acc);
      }
#pragma unroll
      for (int r = 0; r < 8; ++r) {
        const int m = mt * 16 + r + 8 * hi;
        xop[(size_t)m * LL + col] = acc[r];
      }
    }
  }
}

extern "C" void kernel_launch(void* const* d_in, const int* in_sizes, int n_in,
                              void* d_out, int out_size, void* d_ws, size_t ws_size,
                              hipStream_t stream) {
  const float* x   = (const float*)d_in[0];
  const float* W   = (const float*)d_in[1];
  const float* ce  = (const float*)d_in[2];
  const float* ci  = (const float*)d_in[3];
  const float* gd  = (const float*)d_in[4];
  const float* gd2 = (const float*)d_in[5];
  const float* aw  = (const float*)d_in[6];
  float* out = (float*)d_out;
  float* ws  = (float*)d_ws;

  float* gl   = ws + GL_OFF;
  float* gin  = ws + GIN_OFF;
  float* gout = ws + GOUT_OFF;
  float* invq = ws + INVQ_OFF;
  float* invk = ws + INVK_OFF;
  float* dynw = ws + DYNW_OFF;

  colmax_k <<<BB * CC, 256, 0, stream>>>(x, gl);
  gates_k  <<<BB, 256, 0, stream>>>(gl, ce, ci, gd, gd2, gin, gout);
  dynw_k   <<<6144, 256, 0, stream>>>(W, gin, gout, dynw);
  gemm_qkv_k<<<dim3(LL / 128, OCC / 256, BB), 256, 0, stream>>>(dynw, x, out);
  rownorm_k<<<2 * BB * CC, 256, 0, stream>>>(out, invq, invk);
  attn_xo_k<<<BB * HEADS, 256, 0, stream>>>(invq, invk, aw, out);
}